// NodeLabel_34866544508928
// MI455X (gfx1250) — compile-verified
//
#include <hip/hip_runtime.h>

typedef __attribute__((ext_vector_type(2))) float v2f;
typedef __attribute__((ext_vector_type(8))) float v8f;

#define DIM 128

// ---------------------------------------------------------------------------
// Native CDNA5 no-return FP32 atomic add (STOREcnt-tracked, fire-and-forget).
// Guarantees global_atomic_add_f32 instead of a possible CAS loop.
// ---------------------------------------------------------------------------
__device__ __forceinline__ void atomic_fadd(float* p, float v) {
  asm volatile("global_atomic_add_f32 %0, %1, off"
               :
               : "v"(p), "v"(v)
               : "memory");
}

// ---------------------------------------------------------------------------
// Kernel 1: xw = x * node_weight (broadcast over row), zero h1/h2/deg.
// float4 (b128) path; one thread = 4 floats.
// ---------------------------------------------------------------------------
__global__ void nl_init_kernel(const float4* __restrict__ x4,
                               const float* __restrict__ w,
                               float4* __restrict__ xw4,
                               float4* __restrict__ h14,
                               float4* __restrict__ h24,
                               float* __restrict__ deg,
                               int N) {
  long long i = (long long)blockIdx.x * blockDim.x + threadIdx.x;
  long long total4 = (long long)N * (DIM / 4);
  if (i < total4) {
    float4 xv = x4[i];
    float wv = w[i >> 5];  // (i*4)>>7 : 32 float4 per node
    float4 o;
    o.x = xv.x * wv;
    o.y = xv.y * wv;
    o.z = xv.z * wv;
    o.w = xv.w * wv;
    xw4[i] = o;
    float4 z;
    z.x = z.y = z.z = z.w = 0.0f;
    h14[i] = z;
    h24[i] = z;
  }
  if (i < N) deg[i] = 0.0f;
}

// ---------------------------------------------------------------------------
// Kernel 2/3: scatter-add  dst[row] += src[col].  One wave of 32 lanes per
// edge; each lane reads a float4 chunk of the 128-dim source vector and does
// 4 native f32 atomic adds. Lane 0 adds 1.0 to deg[row] on the first pass.
// h1/h2 (25.6 MB each) are L2-resident -> atomics run at L2 rate.
// ---------------------------------------------------------------------------
__global__ void nl_scatter_kernel(const int* __restrict__ ei,
                                  const float* __restrict__ src,
                                  float* __restrict__ dst,
                                  float* __restrict__ deg,
                                  int E, int do_deg) {
  long long t = (long long)blockIdx.x * blockDim.x + threadIdx.x;
  int e = (int)(t >> 5);
  int lane = (int)(t & 31);
  if (e >= E) return;
  int row = ei[e];
  int col = ei[E + e];
  const float4* s =
      reinterpret_cast<const float4*>(src + (size_t)col * DIM) + lane;
  float4 val = *s;
  float* d = dst + (size_t)row * DIM + lane * 4;
  atomic_fadd(d + 0, val.x);
  atomic_fadd(d + 1, val.y);
  atomic_fadd(d + 2, val.z);
  atomic_fadd(d + 3, val.w);
  if (do_deg && lane == 0) atomic_fadd(deg + row, 1.0f);
}

// ---------------------------------------------------------------------------
// Kernel 4: per-query-edge dot products via V_WMMA_F32_16X16X4_F32.
// One wave handles 16 edges. A-tile rows = u-side vectors; B-tile cols =
// v-side vectors; the diagonal of the 16x16 f32 accumulator is the per-edge
// dot. Lane l holds (row/col index = l&15, K-pair = 2*(l>>4)) -> identical
// per-lane layout for A and B operands, so each loaded float2 feeds both.
// ---------------------------------------------------------------------------
__device__ __forceinline__ v8f wmma_f32(v2f a, v2f b, v8f c) {
  return __builtin_amdgcn_wmma_f32_16x16x4_f32(
      /*neg_a=*/false, a, /*neg_b=*/false, b,
      /*c_mod=*/(short)0, c, /*reuse_a=*/false, /*reuse_b=*/false);
}

__device__ __forceinline__ float sel8(v8f c, int r) {
  float x = c[0];
  if (r == 1) x = c[1];
  if (r == 2) x = c[2];
  if (r == 3) x = c[3];
  if (r == 4) x = c[4];
  if (r == 5) x = c[5];
  if (r == 6) x = c[6];
  if (r == 7) x = c[7];
  return x;
}

__global__ void nl_query_kernel(const int* __restrict__ edges,
                                const float* __restrict__ xw,
                                const float* __restrict__ h1,
                                const float* __restrict__ h2,
                                const float* __restrict__ deg,
                                float* __restrict__ out,
                                int EQ) {
  int wave = (int)((blockIdx.x * (long long)blockDim.x + threadIdx.x) >> 5);
  int lane = threadIdx.x & 31;
  int ntiles = EQ >> 4;
  if (wave >= ntiles) return;  // wave-uniform: EXEC stays all-ones for WMMA

  int m = lane & 15;
  int hi = lane >> 4;
  int e = wave * 16 + m;

  int u = edges[e];
  int v = edges[EQ + e];
  float du = deg[u];
  float dv = deg[v];

  const float* pu1 = h1 + (size_t)u * DIM;
  const float* pu2 = h2 + (size_t)u * DIM;
  const float* pux = xw + (size_t)u * DIM;
  const float* pv1 = h1 + (size_t)v * DIM;
  const float* pv2 = h2 + (size_t)v * DIM;
  const float* pvx = xw + (size_t)v * DIM;

  v8f acc11 = {}, acc12 = {}, acc21 = {}, acc22 = {}, accS12 = {}, accS21 = {};

#pragma unroll
  for (int k0 = 0; k0 < DIM; k0 += 4) {
    int o = k0 + 2 * hi;
    v2f a1 = *(const v2f*)(pu1 + o);
    v2f a2 = *(const v2f*)(pu2 + o);
    v2f ax = *(const v2f*)(pux + o);
    v2f b1 = *(const v2f*)(pv1 + o);
    v2f b2 = *(const v2f*)(pv2 + o);
    v2f bx = *(const v2f*)(pvx + o);
    v2f gu = a2 - ax * du;  // h2u - du*xw_u
    v2f gv = b2 - bx * dv;  // h2v - dv*xw_v

    acc11 = wmma_f32(a1, b1, acc11);    // h1u . h1v
    acc12 = wmma_f32(a1, b2, acc12);    // h1u . h2v
    acc21 = wmma_f32(a2, b1, acc21);    // h2u . h1v
    acc22 = wmma_f32(gu, gv, acc22);    // (h2u-du*xwu) . (h2v-dv*xwv)
    accS12 = wmma_f32(a1, a2, accS12);  // h1u . h2u
    accS21 = wmma_f32(b1, b2, accS21);  // h1v . h2v
  }

  // deg outputs (rows 6,7): lanes 0..15 each own one edge
  if (hi == 0) {
    out[6 * (size_t)EQ + e] = du;
    out[7 * (size_t)EQ + e] = dv;
  }

  // Diagonal extraction. C/D layout: VGPR r holds (M=r, N=lane) for lanes
  // 0-15 and (M=r+8, N=lane-16) for lanes 16-31. Diagonal (i,i):
  //   i in 0..7  -> VGPR i,   lane i
  //   i in 8..15 -> VGPR i-8, lane i+16  (lanes 24..31)
  int r = -1, el = 0;
  if (lane < 8) {
    r = lane;
    el = lane;
  } else if (lane >= 24) {
    r = lane - 24;
    el = lane - 16;
  }
  if (r >= 0) {
    int eo = wave * 16 + el;
    out[0 * (size_t)EQ + eo] = sel8(acc11, r);
    out[1 * (size_t)EQ + eo] = sel8(acc12, r);
    out[2 * (size_t)EQ + eo] = sel8(acc21, r);
    out[3 * (size_t)EQ + eo] = sel8(acc22, r);
    out[4 * (size_t)EQ + eo] = sel8(accS12, r);
    out[5 * (size_t)EQ + eo] = sel8(accS21, r);
  }
}

// ---------------------------------------------------------------------------
// Launch
// ---------------------------------------------------------------------------
extern "C" void kernel_launch(void* const* d_in, const int* in_sizes, int n_in,
                              void* d_out, int out_size, void* d_ws,
                              size_t ws_size, hipStream_t stream) {
  const int* edges = (const int*)d_in[0];       // [2, EQ]
  const int* edge_index = (const int*)d_in[1];  // [2, E]
  const float* x = (const float*)d_in[2];       // [N, 128]
  const float* w = (const float*)d_in[3];       // [N, 1]

  int EQ = in_sizes[0] / 2;
  int E = in_sizes[1] / 2;
  int N = in_sizes[3];

  float* wsf = (float*)d_ws;
  float* xw = wsf;                    // N*128
  float* h1 = xw + (size_t)N * DIM;   // N*128
  float* h2 = h1 + (size_t)N * DIM;   // N*128
  float* deg = h2 + (size_t)N * DIM;  // N
  float* out = (float*)d_out;         // [8, EQ]

  {
    long long total4 = (long long)N * (DIM / 4);
    int blocks = (int)((total4 + 255) / 256);
    nl_init_kernel<<<blocks, 256, 0, stream>>>(
        (const float4*)x, w, (float4*)xw, (float4*)h1, (float4*)h2, deg, N);
  }
  {
    long long threads = (long long)E * 32;
    int blocks = (int)((threads + 255) / 256);
    nl_scatter_kernel<<<blocks, 256, 0, stream>>>(edge_index, xw, h1, deg, E, 1);
    nl_scatter_kernel<<<blocks, 256, 0, stream>>>(edge_index, h1, h2, deg, E, 0);
  }
  {
    int ntiles = EQ >> 4;  // 16 edges per wave
    long long threads = (long long)ntiles * 32;
    int blocks = (int)((threads + 255) / 256);
    nl_query_kernel<<<blocks, 256, 0, stream>>>(edges, xw, h1, h2, deg, out, EQ);
  }
}